// KCLWOPosLoss_54855322304749
// MI455X (gfx1250) — compile-verified
//
#include <hip/hip_runtime.h>
#include <hip/hip_bf16.h>
#include <math.h>
#include <stdint.h>

typedef float v2f __attribute__((ext_vector_type(2)));
typedef float v4f __attribute__((ext_vector_type(4)));
typedef float v8f __attribute__((ext_vector_type(8)));

constexpr int Dn = 64;
constexpr int Nn = 4096;
constexpr int dn = 512;
constexpr int KNEG = 64;
constexpr int LP = 2;
constexpr int Pn = Dn - LP;      // 62 pairs
constexpr int TWO_D = 2 * Dn;    // 128 sample rows
constexpr int CAND = TWO_D - 2;  // 126 candidates per pair
constexpr float TEMP_ = 0.1f;
constexpr float EPS_ = 1e-8f;

// ---------------------------------------------------------------- utilities

__device__ __forceinline__ void threefry2x32(uint32_t k0, uint32_t k1,
                                             uint32_t x0, uint32_t x1,
                                             uint32_t* o0, uint32_t* o1) {
  const uint32_t ks0 = k0, ks1 = k1, ks2 = k0 ^ k1 ^ 0x1BD11BDAu;
  x0 += ks0; x1 += ks1;
  #define TF_R4(a, b, c, d)                                      \
    x0 += x1; x1 = (x1 << (a)) | (x1 >> (32 - (a))); x1 ^= x0;   \
    x0 += x1; x1 = (x1 << (b)) | (x1 >> (32 - (b))); x1 ^= x0;   \
    x0 += x1; x1 = (x1 << (c)) | (x1 >> (32 - (c))); x1 ^= x0;   \
    x0 += x1; x1 = (x1 << (d)) | (x1 >> (32 - (d))); x1 ^= x0;
  TF_R4(13, 15, 26, 6)  x0 += ks1; x1 += ks2 + 1u;
  TF_R4(17, 29, 16, 24) x0 += ks2; x1 += ks0 + 2u;
  TF_R4(13, 15, 26, 6)  x0 += ks0; x1 += ks1 + 3u;
  TF_R4(17, 29, 16, 24) x0 += ks1; x1 += ks2 + 4u;
  TF_R4(13, 15, 26, 6)  x0 += ks2; x1 += ks0 + 5u;
  #undef TF_R4
  *o0 = x0; *o1 = x1;
}

__device__ __forceinline__ float logaddexp_f(float a, float b) {
  float mx = fmaxf(a, b), mn = fminf(a, b);
  return mx + log1pf(expf(mn - mx));
}

// ------------------------------------------------- kernel 0: zero sample buf

__global__ void k_zero(float* __restrict__ samples) {
  int base = (blockIdx.x * 256 + threadIdx.x) * 4;  // 64 blocks * 256 thr * 4
  #pragma unroll
  for (int c = 0; c < 4; ++c) samples[base + c] = 0.0f;
}

// --------------------- kernel 1: one streaming pass -> column sums + norms^2
// grid (8, 64): blockIdx.y = D, blockIdx.x = slab of 512 rows.
// 256 threads = 8 waves; each wave owns 64 consecutive rows.
// Lane loads 4x float4 (NT b128) per row; fused row-sqnorm shfl reduce.

__global__ void k_sums_norms(const float* __restrict__ I,
                             float* __restrict__ samples,
                             float* __restrict__ norms2) {
  const int Dd   = blockIdx.y;
  const int slab = blockIdx.x;
  const int wave = threadIdx.x >> 5;
  const int lane = threadIdx.x & 31;

  __shared__ float red[dn];
  for (int t = threadIdx.x; t < dn; t += blockDim.x) red[t] = 0.0f;
  __syncthreads();

  const float* base = I + (size_t)Dd * Nn * dn;
  const int row0 = slab * 512 + wave * 64;

  v4f acc0 = 0.f, acc1 = 0.f, acc2 = 0.f, acc3 = 0.f;

  for (int r = 0; r < 64; ++r) {
    const int row = row0 + r;
    const v4f* rp = (const v4f*)(base + (size_t)row * dn);
    v4f v0 = __builtin_nontemporal_load(rp + lane);
    v4f v1 = __builtin_nontemporal_load(rp + lane + 32);
    v4f v2 = __builtin_nontemporal_load(rp + lane + 64);
    v4f v3 = __builtin_nontemporal_load(rp + lane + 96);
    acc0 += v0; acc1 += v1; acc2 += v2; acc3 += v3;
    float sq = v0.x*v0.x + v0.y*v0.y + v0.z*v0.z + v0.w*v0.w
             + v1.x*v1.x + v1.y*v1.y + v1.z*v1.z + v1.w*v1.w
             + v2.x*v2.x + v2.y*v2.y + v2.z*v2.z + v2.w*v2.w
             + v3.x*v3.x + v3.y*v3.y + v3.z*v3.z + v3.w*v3.w;
    #pragma unroll
    for (int m = 16; m > 0; m >>= 1) sq += __shfl_xor(sq, m, 32);
    if (lane == 0) norms2[Dd * Nn + row] = sq;
  }

  // wave-private column sums -> LDS (ds_add_f32), then one global atomic pass
  const int c0 = lane * 4;
  #pragma unroll
  for (int c = 0; c < 4; ++c) {
    atomicAdd(&red[0   + c0 + c], acc0[c]);
    atomicAdd(&red[128 + c0 + c], acc1[c]);
    atomicAdd(&red[256 + c0 + c], acc2[c]);
    atomicAdd(&red[384 + c0 + c], acc3[c]);
  }
  __syncthreads();
  for (int t = threadIdx.x; t < dn; t += blockDim.x)
    atomicAdd(&samples[Dd * dn + t], red[t]);
}

// ----------------------- kernel 2: iterative top-64 argmax per D (in LDS)

__global__ void k_topk(const float* __restrict__ norms2, int* __restrict__ topk) {
  const int Dd = blockIdx.x;
  const int tid = threadIdx.x;
  __shared__ float vals[Nn];
  __shared__ float bval[256];
  __shared__ int   bidx[256];

  for (int i = tid; i < Nn; i += 256) vals[i] = norms2[Dd * Nn + i];
  __syncthreads();

  for (int k = 0; k < KNEG; ++k) {
    float best = -1.0f; int bi = Nn;
    for (int i = tid; i < Nn; i += 256) {
      float v = vals[i];
      if (v > best) { best = v; bi = i; }   // ascending scan => smallest idx on tie
    }
    bval[tid] = best; bidx[tid] = bi;
    __syncthreads();
    for (int s = 128; s > 0; s >>= 1) {
      if (tid < s) {
        float ov = bval[tid + s]; int oi = bidx[tid + s];
        if (ov > bval[tid] || (ov == bval[tid] && oi < bidx[tid])) {
          bval[tid] = ov; bidx[tid] = oi;
        }
      }
      __syncthreads();
    }
    if (tid == 0) { topk[Dd * KNEG + k] = bidx[0]; vals[bidx[0]] = -1.0f; }
    __syncthreads();
  }
}

// --------------- kernel 3: sums_neg = sums - sum(top-64 rows)  -> samples[64+]

__global__ void k_sumneg(const float* __restrict__ I,
                         const int* __restrict__ topk,
                         float* __restrict__ samples) {
  const int Dd = blockIdx.x;
  const int tid = threadIdx.x;
  __shared__ int idx[KNEG];
  if (tid < KNEG) idx[tid] = topk[Dd * KNEG + tid];
  __syncthreads();

  const float* base = I + (size_t)Dd * Nn * dn;
  for (int c = tid; c < dn; c += 256) {
    float acc = samples[Dd * dn + c];
    #pragma unroll 4
    for (int k = 0; k < KNEG; ++k) acc -= base[(size_t)idx[k] * dn + c];
    samples[(Dn + Dd) * dn + c] = acc;
  }
}

// ------------- kernel 4: Gram = S * S^T  (128x128, k=512) via WMMA f32 16x16x4
// 64 blocks = 8x8 tiles, one wave each. f32 A/B/C -> exact fp32 dot products.

__global__ void k_gram(const float* __restrict__ S, float* __restrict__ G) {
  const int tm = blockIdx.x >> 3;
  const int tn = blockIdx.x & 7;
  const int lane = threadIdx.x;
  const int m    = lane & 15;
  const int koff = (lane >> 4) * 2;          // lanes 0-15: K{0,1}; 16-31: K{2,3}

  const float* pa = S + (size_t)(tm * 16 + m) * dn + koff;
  const float* pb = S + (size_t)(tn * 16 + m) * dn + koff;

  v8f c = {};
  for (int kk = 0; kk < dn; kk += 4) {
    v2f a = *(const v2f*)(pa + kk);
    v2f b = *(const v2f*)(pb + kk);
    c = __builtin_amdgcn_wmma_f32_16x16x4_f32(false, a, false, b,
                                              (short)0, c, false, false);
  }

  const int n  = lane & 15;
  const int mb = (lane >> 4) * 8;            // C layout: VGPR r = M=r / M=8+r
  #pragma unroll
  for (int r = 0; r < 8; ++r)
    G[(size_t)(tm * 16 + mb + r) * TWO_D + tn * 16 + n] = c[r];
}

// ------------- kernel 5: threefry selection + cosine sims + logsumexp + loss
// Single block, 128 threads. Thread p < 62 handles pair (p, p+2).

__global__ void k_loss(const float* __restrict__ g1, const float* __restrict__ g2,
                       const float* __restrict__ G, float* __restrict__ out) {
  const int tid = threadIdx.x;

  __shared__ uint32_t skeys[Pn][CAND];   // 62*126*4 = 31.2 KB
  __shared__ short    selbuf[Pn][KNEG];  // 7.9 KB
  __shared__ float    nrm[TWO_D];        // eps-clamped row norms
  __shared__ float    r0[128], r1[128], r2[128];
  __shared__ float    pos_s;

  // ---- pos = cos(g1, g2) / T
  float dv = 0.f, n1 = 0.f, n2 = 0.f;
  for (int i = tid; i < dn; i += 128) {
    float a = g1[i], b = g2[i];
    dv += a * b; n1 += a * a; n2 += b * b;
  }
  r0[tid] = dv; r1[tid] = n1; r2[tid] = n2;
  __syncthreads();
  for (int s = 64; s > 0; s >>= 1) {
    if (tid < s) { r0[tid] += r0[tid+s]; r1[tid] += r1[tid+s]; r2[tid] += r2[tid+s]; }
    __syncthreads();
  }
  if (tid == 0)
    pos_s = (r0[0] / (fmaxf(sqrtf(r1[0]), EPS_) * fmaxf(sqrtf(r2[0]), EPS_))) / TEMP_;

  // ---- row norms from Gram diagonal
  if (tid < TWO_D) nrm[tid] = fmaxf(sqrtf(G[(size_t)tid * TWO_D + tid]), EPS_);
  __syncthreads();

  float loss = 0.0f;
  if (tid < Pn) {
    const int p = tid;

    // key_p from jax.random.split(key(42), 62): word w of 124-word counter run
    auto word = [](int w) -> uint32_t {
      uint32_t a, b;
      if (w < 62) { threefry2x32(0u, 42u, (uint32_t)w, (uint32_t)(w + 62), &a, &b); return a; }
      threefry2x32(0u, 42u, (uint32_t)(w - 62), (uint32_t)w, &a, &b); return b;
    };
    const uint32_t kp0 = word(2 * p);
    const uint32_t kp1 = word(2 * p + 1);

    // key, subkey = split(key_p): counts [0,1,2,3] -> lanes (0,2),(1,3); subkey = row 1
    uint32_t y0, y1, z0, z1;
    threefry2x32(kp0, kp1, 0u, 2u, &y0, &y1);
    threefry2x32(kp0, kp1, 1u, 3u, &z0, &z1);
    const uint32_t sub0 = y1, sub1 = z1;

    // sort keys for n=126: lanes (i, i+63)
    for (int i = 0; i < 63; ++i) {
      uint32_t a, b;
      threefry2x32(sub0, sub1, (uint32_t)i, (uint32_t)(i + 63), &a, &b);
      skeys[p][i] = a; skeys[p][i + 63] = b;
    }

    // stable ascending rank == position after lax.sort_key_val; keep ranks < 64
    const int ii = p, jj = p + LP;
    for (int t = 0; t < CAND; ++t) {
      const uint32_t kt = skeys[p][t];
      int rank = 0;
      for (int u = 0; u < CAND; ++u) {
        const uint32_t ku = skeys[p][u];
        rank += (ku < kt) || (ku == kt && u < t);
      }
      if (rank < KNEG) {
        int c = t + (t >= ii);
        c += (c >= jj);
        selbuf[p][rank] = (short)c;
      }
    }

    // sims + logsumexp (two passes: max, then sum-exp)
    const float ni = nrm[ii], nj = nrm[jj];
    float mi = -INFINITY, mj = -INFINITY;
    for (int k = 0; k < KNEG; ++k) {
      const int c = selbuf[p][k];
      const float nc = nrm[c];
      mi = fmaxf(mi, G[(size_t)ii * TWO_D + c] / (ni * nc) / TEMP_);
      mj = fmaxf(mj, G[(size_t)jj * TWO_D + c] / (nj * nc) / TEMP_);
    }
    float si = 0.f, sj = 0.f;
    for (int k = 0; k < KNEG; ++k) {
      const int c = selbuf[p][k];
      const float nc = nrm[c];
      si += expf(G[(size_t)ii * TWO_D + c] / (ni * nc) / TEMP_ - mi);
      sj += expf(G[(size_t)jj * TWO_D + c] / (nj * nc) / TEMP_ - mj);
    }
    const float lse_i = mi + logf(si);
    const float lse_j = mj + logf(sj);
    const float pos = pos_s;
    loss = (logaddexp_f(pos, lse_i) - pos) + (logaddexp_f(pos, lse_j) - pos);
  }

  r0[tid] = loss;
  __syncthreads();
  for (int s = 64; s > 0; s >>= 1) {
    if (tid < s) r0[tid] += r0[tid + s];
    __syncthreads();
  }
  if (tid == 0) out[0] = r0[0];
}

// ---------------------------------------------------------------- launcher

extern "C" void kernel_launch(void* const* d_in, const int* in_sizes, int n_in,
                              void* d_out, int out_size, void* d_ws, size_t ws_size,
                              hipStream_t stream) {
  const float* I  = (const float*)d_in[0];
  // g0 (d_in[1]) is unused by the reference
  const float* g1 = (const float*)d_in[2];
  const float* g2 = (const float*)d_in[3];
  float* out = (float*)d_out;

  float* samples = (float*)d_ws;                       // [128][512]
  float* norms2  = samples + TWO_D * dn;               // [64][4096]
  int*   topk    = (int*)(norms2 + Dn * Nn);           // [64][64]
  float* G       = (float*)(topk + Dn * KNEG);         // [128][128]

  k_zero      <<<dim3(64),    dim3(256), 0, stream>>>(samples);
  k_sums_norms<<<dim3(8, 64), dim3(256), 0, stream>>>(I, samples, norms2);
  k_topk      <<<dim3(Dn),    dim3(256), 0, stream>>>(norms2, topk);
  k_sumneg    <<<dim3(Dn),    dim3(256), 0, stream>>>(I, topk, samples);
  k_gram      <<<dim3(64),    dim3(32),  0, stream>>>(samples, G);
  k_loss      <<<dim3(1),     dim3(128), 0, stream>>>(g1, g2, G, out);
}